// ROIHead_22557168238613
// MI455X (gfx1250) — compile-verified
//
#include <hip/hip_runtime.h>
#include <hip/hip_bf16.h>

// ---------------------------------------------------------------------------
// ROI head for MI455X (gfx1250, wave32).
// FC6/FC7 run on the matrix pipe via V_WMMA_F32_16X16X4_F32 (fp32 kept for
// numerical fidelity: the pipeline ends in argmax-driven NMS which is
// discrete and error-amplifying). Working set (~110MB) is L2-resident.
// Each wave owns TWO 16x16 accumulator tiles -> two independent WMMA
// dependency chains (hides XDL latency) and 2x reuse of the B operand.
// ---------------------------------------------------------------------------

typedef __attribute__((ext_vector_type(2))) float v2f;
typedef __attribute__((ext_vector_type(8))) float v8f;

#define NPROP   1000
#define CIN     256
#define FH      50
#define FW      50
#define POOLN   7
#define KBIN    10
#define FCDIM   1024
#define K6      (CIN * POOLN * POOLN)   // 12544
#define NCLS    21
#define NCAND   (NPROP * (NCLS - 1))    // 20000
#define TOPK    100
#define MAX_DWH 4.1351666f              // log(1000/16)
#define NEGF    -1e30f

// ---------------------------------------------------------------------------
// 1) ROI max pooling (replicates jnp reference bit-for-bit where possible)
// ---------------------------------------------------------------------------
__global__ __launch_bounds__(256) void roi_pool_kernel(
    const float* __restrict__ feat,       // [CIN][FH][FW]
    const float* __restrict__ rois,       // [NPROP][4]
    float* __restrict__ X)                // [NPROP][K6]
{
    __shared__ int hs_[POOLN], he_[POOLN], ws_[POOLN], we_[POOLN];
    const int p = blockIdx.x;
    const float* roi = rois + p * 4;
    // jnp.round == round-to-nearest-even == rintf
    const int x1 = (int)rintf(roi[0] * 0.0625f);
    const int y1 = (int)rintf(roi[1] * 0.0625f);
    const int x2 = (int)rintf(roi[2] * 0.0625f);
    const int y2 = (int)rintf(roi[3] * 0.0625f);
    const float rw = (float)max(x2 - x1 + 1, 1);
    const float rh = (float)max(y2 - y1 + 1, 1);
    const float bh = rh / 7.0f;
    const float bw = rw / 7.0f;
    if (threadIdx.x < POOLN) {
        const int i = threadIdx.x;
        const float pr = (float)i;
        hs_[i] = min(max((int)floorf(pr * bh) + y1, 0), FH);
        he_[i] = min(max((int)ceilf((pr + 1.0f) * bh) + y1, 0), FH);
        ws_[i] = min(max((int)floorf(pr * bw) + x1, 0), FW);
        we_[i] = min(max((int)ceilf((pr + 1.0f) * bw) + x1, 0), FW);
    }
    __syncthreads();
    float* xrow = X + (size_t)p * K6;
    for (int o = threadIdx.x; o < K6; o += 256) {
        const int c  = o / 49;
        const int r  = o % 49;
        const int ph = r / 7;
        const int pw = r % 7;
        const int h0 = hs_[ph], h1 = he_[ph];
        const int w0 = ws_[pw], w1 = we_[pw];
        float m;
        if (h1 <= h0 || w1 <= w0) {
            m = 0.0f;                        // "empty" bin -> 0
        } else {
            m = NEGF;
            const int hlim = min(h1, h0 + KBIN);   // reference caps window at KBIN
            const int wlim = min(w1, w0 + KBIN);
            const float* fc = feat + (size_t)c * (FH * FW);
            for (int h = h0; h < hlim; ++h)
                for (int w = w0; w < wlim; ++w)
                    m = fmaxf(m, fc[h * FW + w]);
        }
        xrow[o] = m;
    }
}

// ---------------------------------------------------------------------------
// 2) Fully-connected layer: out[M][N] = relu?( A[M][K] @ W[N][K]^T + b[N] )
//    4 waves / block; each wave owns TWO 16x16 WMMA tiles (rows i0..i0+31),
//    block covers 32(M) x 64(N). A chunk staged in LDS (stride 66 ->
//    conflict-free column reads, 8B-aligned v2f operand loads).
//    fp32 WMMA 16x16x4; two accumulator chains per wave; B loaded once,
//    used by both tiles; next B chunk prefetched (global_prefetch).
// ---------------------------------------------------------------------------
#define KCHUNK 64
#define LDSA_S 66
#define MROWS  32     // rows per block (2 x 16-row WMMA tiles)

__global__ __launch_bounds__(128) void fc_wmma_kernel(
    const float* __restrict__ A, const float* __restrict__ W,
    const float* __restrict__ bias, float* __restrict__ out,
    int M, int N, int K, int relu)
{
    __shared__ float lA[MROWS * LDSA_S];
    const int i0   = blockIdx.x * MROWS;
    const int wave = threadIdx.x >> 5;
    const int lane = threadIdx.x & 31;
    const int j0   = blockIdx.y * 64 + wave * 16;
    const int lm   = lane & 15;   // matrix row (A) / col (B,D)
    const int lh   = lane >> 4;   // K-pair select (A,B) / +8 row (D)

    v8f acc0 = {};
    v8f acc1 = {};
    const float* wrow = W + (size_t)(j0 + lm) * K;

    for (int kb = 0; kb < K; kb += KCHUNK) {
        // prefetch next B chunk for this lane's weight row
        if (kb + KCHUNK < K)
            __builtin_prefetch(&wrow[kb + KCHUNK], 0, 0);
        // cooperative stage of A[i0:i0+32][kb:kb+64] into LDS
        for (int t = threadIdx.x; t < MROWS * KCHUNK; t += 128) {
            const int r = t >> 6;
            const int c = t & 63;
            const int gr = i0 + r;
            lA[r * LDSA_S + c] = (gr < M) ? A[(size_t)gr * K + kb + c] : 0.0f;
        }
        __syncthreads();
        #pragma unroll
        for (int kk = 0; kk < KCHUNK; kk += 4) {
            const int ksub = kk + lh * 2;
            // B 4x16 layout: lane(lm, half lh) holds B[lh*2..+1][lm] = W[j0+lm][k..k+1]
            v2f b = *(const v2f*)&wrow[kb + ksub];
            // A 16x4 layout: lane(lm, half lh) holds A[row][lh*2 .. lh*2+1]
            v2f a0 = *(const v2f*)&lA[lm * LDSA_S + ksub];
            v2f a1 = *(const v2f*)&lA[(lm + 16) * LDSA_S + ksub];
            acc0 = __builtin_amdgcn_wmma_f32_16x16x4_f32(
                false, a0, false, b, (short)0, acc0, false, false);
            acc1 = __builtin_amdgcn_wmma_f32_16x16x4_f32(
                false, a1, false, b, (short)0, acc1, false, false);
        }
        __syncthreads();
    }

    // D 16x16 f32 layout: vgpr v, lane(lm,lh) -> row base + lh*8 + v, col j0 + lm
    const int col = j0 + lm;
    const float bv = bias[col];
    #pragma unroll
    for (int vg = 0; vg < 8; ++vg) {
        const int row0 = i0 + lh * 8 + vg;
        if (row0 < M) {
            float v = acc0[vg] + bv;
            if (relu) v = fmaxf(v, 0.0f);
            out[(size_t)row0 * N + col] = v;
        }
        const int row1 = i0 + 16 + lh * 8 + vg;
        if (row1 < M) {
            float v = acc1[vg] + bv;
            if (relu) v = fmaxf(v, 0.0f);
            out[(size_t)row1 * N + col] = v;
        }
    }
}

// ---------------------------------------------------------------------------
// 3) cls + bbox heads: 105 dots of length 1024 per proposal
// ---------------------------------------------------------------------------
__global__ __launch_bounds__(128) void heads_kernel(
    const float* __restrict__ H7,
    const float* __restrict__ cls_w,  const float* __restrict__ cls_b,
    const float* __restrict__ bbox_w, const float* __restrict__ bbox_b,
    float* __restrict__ CLS, float* __restrict__ BT)
{
    __shared__ float h[FCDIM];
    const int p = blockIdx.x;
    const float* hrow = H7 + (size_t)p * FCDIM;
    for (int t = threadIdx.x; t < FCDIM; t += 128) h[t] = hrow[t];
    __syncthreads();
    const int o = threadIdx.x;
    if (o < NCLS) {
        const float* wr = cls_w + (size_t)o * FCDIM;
        float s = cls_b[o];
        for (int k = 0; k < FCDIM; ++k) s += h[k] * wr[k];
        CLS[p * NCLS + o] = s;
    } else if (o < NCLS + 84) {
        const int q = o - NCLS;
        const float* wr = bbox_w + (size_t)q * FCDIM;
        float s = bbox_b[q];
        for (int k = 0; k < FCDIM; ++k) s += h[k] * wr[k];
        BT[p * 84 + q] = s;
    }
}

// ---------------------------------------------------------------------------
// 4) softmax + box decode + candidate prep (classes 1..20)
// ---------------------------------------------------------------------------
__global__ __launch_bounds__(32) void decode_kernel(
    const float* __restrict__ CLS, const float* __restrict__ BT,
    const float* __restrict__ rois,
    const int* __restrict__ ih_p, const int* __restrict__ iw_p,
    float* __restrict__ CB, float* __restrict__ CBS, float* __restrict__ CS,
    float* __restrict__ WORK, float* __restrict__ AREA, float* __restrict__ LBL)
{
    const int p = blockIdx.x;
    const float imh = (float)(*ih_p);
    const float imw = (float)(*iw_p);
    const float offscale = fmaxf(imh, imw) + 2.0f;

    const float* s = CLS + p * NCLS;
    float mx = s[0];
    for (int i = 1; i < NCLS; ++i) mx = fmaxf(mx, s[i]);
    float den = 0.0f;
    for (int i = 0; i < NCLS; ++i) den += expf(s[i] - mx);

    const int c = threadIdx.x;
    if (c >= 1 && c < NCLS) {
        const float sc = expf(s[c] - mx) / den;
        const float* roi = rois + p * 4;
        const float w  = roi[2] - roi[0];
        const float h  = roi[3] - roi[1];
        const float cx = roi[0] + 0.5f * w;
        const float cy = roi[1] + 0.5f * h;
        const float* bt = BT + p * 84 + c * 4;
        const float dx = bt[0], dy = bt[1];
        const float dw = fminf(bt[2], MAX_DWH);
        const float dh = fminf(bt[3], MAX_DWH);
        const float pcx = dx * w + cx;
        const float pcy = dy * h + cy;
        const float pww = expf(dw) * w;
        const float phh = expf(dh) * h;
        const float bx1 = fminf(fmaxf(pcx - 0.5f * pww, 0.0f), imw);
        const float by1 = fminf(fmaxf(pcy - 0.5f * phh, 0.0f), imh);
        const float bx2 = fminf(fmaxf(pcx + 0.5f * pww, 0.0f), imw);
        const float by2 = fminf(fmaxf(pcy + 0.5f * phh, 0.0f), imh);
        const int idx = p * (NCLS - 1) + (c - 1);
        CB[idx * 4 + 0] = bx1; CB[idx * 4 + 1] = by1;
        CB[idx * 4 + 2] = bx2; CB[idx * 4 + 3] = by2;
        const float wsz = bx2 - bx1, hsz = by2 - by1;
        const bool valid = (sc > 0.05f) && (wsz >= 1.0f) && (hsz >= 1.0f);
        CS[idx]   = sc;
        WORK[idx] = valid ? sc : -1.0f;
        const float off = (float)c * offscale;     // per-class offset trick
        const float sx1 = bx1 + off, sy1 = by1 + off;
        const float sx2 = bx2 + off, sy2 = by2 + off;
        CBS[idx * 4 + 0] = sx1; CBS[idx * 4 + 1] = sy1;
        CBS[idx * 4 + 2] = sx2; CBS[idx * 4 + 3] = sy2;
        AREA[idx] = (sx2 - sx1) * (sy2 - sy1);
        LBL[idx]  = (float)c;
    }
}

// ---------------------------------------------------------------------------
// 5) greedy NMS: single 1024-thread (32-wave) workgroup, LDS tree argmax
//    with jnp.argmax first-occurrence tie-breaking.
// ---------------------------------------------------------------------------
__global__ __launch_bounds__(1024) void nms_kernel(
    const float* __restrict__ CB, const float* __restrict__ CBS,
    const float* __restrict__ CS, float* __restrict__ WORK,
    const float* __restrict__ AREA, const float* __restrict__ LBL,
    float* __restrict__ out)
{
    __shared__ float rv[1024];
    __shared__ int   ri[1024];
    __shared__ float sb[5];     // selected shifted box + area
    __shared__ int   sj;
    __shared__ float svv;
    const int tid = threadIdx.x;

    for (int t = 0; t < TOPK; ++t) {
        // strided argmax (ascending stripe + strict '>' => first max in stripe)
        float bv = -3.0e38f; int bi = 0x7fffffff;
        for (int i = tid; i < NCAND; i += 1024) {
            const float v = WORK[i];
            if (v > bv) { bv = v; bi = i; }
        }
        rv[tid] = bv; ri[tid] = bi;
        __syncthreads();
        for (int s2 = 512; s2 > 0; s2 >>= 1) {
            if (tid < s2) {
                const float ov = rv[tid + s2]; const int oi = ri[tid + s2];
                if (ov > rv[tid] || (ov == rv[tid] && oi < ri[tid])) {
                    rv[tid] = ov; ri[tid] = oi;
                }
            }
            __syncthreads();
        }
        if (tid == 0) {
            sj = ri[0]; svv = rv[0];
            const int j = sj;
            sb[0] = CBS[j * 4 + 0]; sb[1] = CBS[j * 4 + 1];
            sb[2] = CBS[j * 4 + 2]; sb[3] = CBS[j * 4 + 3];
            sb[4] = AREA[j];
        }
        __syncthreads();
        const int j = sj;
        for (int i = tid; i < NCAND; i += 1024) {
            const float xl = fmaxf(sb[0], CBS[i * 4 + 0]);
            const float yt = fmaxf(sb[1], CBS[i * 4 + 1]);
            const float xr = fminf(sb[2], CBS[i * 4 + 2]);
            const float yb = fminf(sb[3], CBS[i * 4 + 3]);
            const float inter = fmaxf(xr - xl, 0.0f) * fmaxf(yb - yt, 0.0f);
            const float iou = inter / (sb[4] + AREA[i] - inter);
            if (iou > 0.5f || i == j) WORK[i] = -1.0f;
        }
        if (tid == 0) {
            const bool good = svv > 0.0f;
            out[t * 4 + 0] = good ? CB[j * 4 + 0] : 0.0f;
            out[t * 4 + 1] = good ? CB[j * 4 + 1] : 0.0f;
            out[t * 4 + 2] = good ? CB[j * 4 + 2] : 0.0f;
            out[t * 4 + 3] = good ? CB[j * 4 + 3] : 0.0f;
            out[4 * TOPK + t] = good ? CS[j] : 0.0f;
            out[5 * TOPK + t] = good ? LBL[j] : -1.0f;
        }
        __syncthreads();
    }
}

// ---------------------------------------------------------------------------
// launch
// ---------------------------------------------------------------------------
extern "C" void kernel_launch(void* const* d_in, const int* in_sizes, int n_in,
                              void* d_out, int out_size, void* d_ws, size_t ws_size,
                              hipStream_t stream) {
    (void)in_sizes; (void)n_in; (void)out_size; (void)ws_size;
    const float* feat      = (const float*)d_in[0];
    const float* proposals = (const float*)d_in[1];
    const float* fc6_w     = (const float*)d_in[2];
    const float* fc6_b     = (const float*)d_in[3];
    const float* fc7_w     = (const float*)d_in[4];
    const float* fc7_b     = (const float*)d_in[5];
    const float* cls_w     = (const float*)d_in[6];
    const float* cls_b     = (const float*)d_in[7];
    const float* bbox_w    = (const float*)d_in[8];
    const float* bbox_b    = (const float*)d_in[9];
    const int*   image_h   = (const int*)d_in[10];
    const int*   image_w   = (const int*)d_in[11];

    float* ws = (float*)d_ws;
    float* X    = ws;                                    // 1000*12544
    float* H6   = X    + (size_t)NPROP * K6;             // 1000*1024
    float* H7   = H6   + (size_t)NPROP * FCDIM;          // 1000*1024
    float* CLS  = H7   + (size_t)NPROP * FCDIM;          // 1000*21
    float* BT   = CLS  + (size_t)NPROP * NCLS;           // 1000*84
    float* CB   = BT   + (size_t)NPROP * 84;             // 20000*4
    float* CBS  = CB   + (size_t)NCAND * 4;              // 20000*4
    float* CS   = CBS  + (size_t)NCAND * 4;              // 20000
    float* WORK = CS   + (size_t)NCAND;                  // 20000
    float* AREA = WORK + (size_t)NCAND;                  // 20000
    float* LBL  = AREA + (size_t)NCAND;                  // 20000

    roi_pool_kernel<<<NPROP, 256, 0, stream>>>(feat, proposals, X);

    dim3 g6((NPROP + MROWS - 1) / MROWS, FCDIM / 64);    // 32 x 16
    fc_wmma_kernel<<<g6, 128, 0, stream>>>(X,  fc6_w, fc6_b, H6, NPROP, FCDIM, K6,    1);
    fc_wmma_kernel<<<g6, 128, 0, stream>>>(H6, fc7_w, fc7_b, H7, NPROP, FCDIM, FCDIM, 1);

    heads_kernel<<<NPROP, 128, 0, stream>>>(H7, cls_w, cls_b, bbox_w, bbox_b, CLS, BT);
    decode_kernel<<<NPROP, 32, 0, stream>>>(CLS, BT, proposals, image_h, image_w,
                                            CB, CBS, CS, WORK, AREA, LBL);
    nms_kernel<<<1, 1024, 0, stream>>>(CB, CBS, CS, WORK, AREA, LBL, (float*)d_out);
}